// ContextualAttention_23527830847888
// MI455X (gfx1250) — compile-verified
//
#include <hip/hip_runtime.h>

// ---------------------------------------------------------------------------
// Contextual attention for MI455X (gfx1250): two big fp32 GEMMs on
// V_WMMA_F32_16X16X4_F32 with 64x64 wave tiles (16 WMMA per k-step : 8 LDS b64).
// B=4, C=64, H=W=48 -> N=2304 pixels/patches, K=C*9=576
// ---------------------------------------------------------------------------

typedef __attribute__((ext_vector_type(2))) float v2f;
typedef __attribute__((ext_vector_type(8))) float v8f;

#define NPIX 2304   // H*W
#define KDIM 576    // C*9
#define HW   48
#define CCH  64

// D = A(16x4) * B(4x16) + C, full fp32
#define WMMA_F32(a, b, c) \
  __builtin_amdgcn_wmma_f32_16x16x4_f32(false, (a), false, (b), (short)0, (c), false, false)

// ---------------------------------------------------------------------------
// P1: per background patch n: reciprocal L2 norm over (C,3,3), and mask gate
// ---------------------------------------------------------------------------
__global__ __launch_bounds__(256) void prep_norm_kernel(
    const float* __restrict__ bg,    // [64,48,48] (this batch)
    const float* __restrict__ mask,  // [48,48]
    float* __restrict__ mp,          // [2304]
    float* __restrict__ rnrm)        // [2304]
{
  int n = blockIdx.x * 256 + threadIdx.x;
  if (n >= NPIX) return;
  int ny = n / HW, nx = n % HW;

  float msum = 0.f;
  #pragma unroll
  for (int dy = 0; dy < 3; ++dy) {
    int y = ny + dy - 1;
    if ((unsigned)y >= (unsigned)HW) continue;
    #pragma unroll
    for (int dx = 0; dx < 3; ++dx) {
      int x = nx + dx - 1;
      if ((unsigned)x >= (unsigned)HW) continue;
      msum += mask[y * HW + x];
    }
  }
  mp[n] = (msum != 0.f) ? 1.f : 0.f;

  float sq = 0.f;
  for (int c = 0; c < CCH; ++c) {
    const float* bc = bg + c * NPIX;
    #pragma unroll
    for (int dy = 0; dy < 3; ++dy) {
      int y = ny + dy - 1;
      if ((unsigned)y >= (unsigned)HW) continue;
      #pragma unroll
      for (int dx = 0; dx < 3; ++dx) {
        int x = nx + dx - 1;
        if ((unsigned)x >= (unsigned)HW) continue;
        float v = bc[y * HW + x];
        sq += v * v;
      }
    }
  }
  rnrm[n] = 1.f / fmaxf(sqrtf(sq), 1e-4f);
}

// ---------------------------------------------------------------------------
// P2: im2col, n-major layout [n][k] (k fast) for Pf and Pbn
// ---------------------------------------------------------------------------
__global__ __launch_bounds__(256) void prep_unfold_kernel(
    const float* __restrict__ fg, const float* __restrict__ bg,
    const float* __restrict__ rnrm,
    float* __restrict__ Pf, float* __restrict__ Pbn)
{
  int t = blockIdx.x * 256 + threadIdx.x;
  if (t >= NPIX * KDIM) return;
  int n = t / KDIM, k = t % KDIM;
  int c = k / 9, r = k % 9, dy = r / 3, dx = r % 3;
  int y = n / HW + dy - 1, x = n % HW + dx - 1;
  bool in = ((unsigned)y < (unsigned)HW) && ((unsigned)x < (unsigned)HW);
  float vf = 0.f, vb = 0.f;
  if (in) {
    int off = c * NPIX + y * HW + x;
    vf = fg[off];
    vb = bg[off];
  }
  Pf[t]  = vf;
  Pbn[t] = vb * rnrm[n];
}

// ---------------------------------------------------------------------------
// P3: k-major layout [k][n] (n fast) for normalized background (GEMM2 A side)
// ---------------------------------------------------------------------------
__global__ __launch_bounds__(256) void prep_unfold_t_kernel(
    const float* __restrict__ bg, const float* __restrict__ rnrm,
    float* __restrict__ Pbnt)
{
  int t = blockIdx.x * 256 + threadIdx.x;   // t = k*NPIX + n (coalesced writes)
  if (t >= NPIX * KDIM) return;
  int k = t / NPIX, n = t % NPIX;
  int c = k / 9, r = k % 9, dy = r / 3, dx = r % 3;
  int y = n / HW + dy - 1, x = n % HW + dx - 1;
  bool in = ((unsigned)y < (unsigned)HW) && ((unsigned)x < (unsigned)HW);
  float vb = in ? bg[c * NPIX + y * HW + x] : 0.f;
  Pbnt[t] = vb * rnrm[n];
}

// ---------------------------------------------------------------------------
// GEMM1: attn[p][n] = sum_k Pf[p][k] * Pbn[n][k]
// Block tile 128(p) x 256(n), 8 waves (2x4), each wave a 64x64 WMMA tile
// (16 accumulators). K staged in LDS chunks of 32, pitch 36 (conflict-free).
// ---------------------------------------------------------------------------
__global__ __launch_bounds__(256) void gemm1_kernel(
    const float* __restrict__ Pf,   // [NPIX][KDIM]
    const float* __restrict__ Pbn,  // [NPIX][KDIM]
    float* __restrict__ attn)       // [NPIX(p)][NPIX(n)]
{
  __shared__ float la[128][36];
  __shared__ float lb[256][36];

  const int tid = threadIdx.x, lane = tid & 31, wave = tid >> 5;
  const int wp = wave >> 2;       // 0..1 : p sub-tile (64)
  const int wn = wave & 3;        // 0..3 : n sub-tile (64)
  const int p0 = blockIdx.y * 128, n0 = blockIdx.x * 256;
  const int fr = lane & 15;               // fragment row (M or N index)
  const int fk = (lane >> 4) << 1;        // k pair: 0 or 2

  v8f acc[4][4];
  #pragma unroll
  for (int i = 0; i < 4; ++i)
    #pragma unroll
    for (int j = 0; j < 4; ++j) acc[i][j] = (v8f){};

  for (int kc = 0; kc < KDIM; kc += 32) {
    __syncthreads();
    #pragma unroll
    for (int i = 0; i < 4; ++i) {                 // A: 128x32 = 1024 float4
      int e = tid + i * 256;
      int row = e >> 3, col4 = (e & 7) << 2;
      *(float4*)&la[row][col4] =
          *(const float4*)&Pf[(size_t)(p0 + row) * KDIM + kc + col4];
    }
    #pragma unroll
    for (int i = 0; i < 8; ++i) {                 // B: 256x32 = 2048 float4
      int e = tid + i * 256;
      int row = e >> 3, col4 = (e & 7) << 2;
      *(float4*)&lb[row][col4] =
          *(const float4*)&Pbn[(size_t)(n0 + row) * KDIM + kc + col4];
    }
    __syncthreads();

    #pragma unroll
    for (int kk = 0; kk < 32; kk += 4) {
      v2f af[4], bf[4];
      #pragma unroll
      for (int i = 0; i < 4; ++i)
        af[i] = *(const v2f*)&la[wp * 64 + i * 16 + fr][kk + fk];
      #pragma unroll
      for (int j = 0; j < 4; ++j)
        bf[j] = *(const v2f*)&lb[wn * 64 + j * 16 + fr][kk + fk];
      #pragma unroll
      for (int i = 0; i < 4; ++i)
        #pragma unroll
        for (int j = 0; j < 4; ++j)
          acc[i][j] = WMMA_F32(af[i], bf[j], acc[i][j]);
    }
  }

  // D layout: VGPR r, lane l -> row = r + 8*(l>>4), col = l&15
  const int dr = (lane >> 4) << 3, dc = lane & 15;
  #pragma unroll
  for (int i = 0; i < 4; ++i)
    #pragma unroll
    for (int j = 0; j < 4; ++j) {
      float* dst = attn + (size_t)(p0 + wp * 64 + i * 16 + dr) * NPIX
                        + n0 + wn * 64 + j * 16 + dc;
      #pragma unroll
      for (int r = 0; r < 8; ++r) dst[(size_t)r * NPIX] = acc[i][j][r];
    }
}

// ---------------------------------------------------------------------------
// Softmax over n per row p, gated by mp, scale LAMBDA=10 (online m/s + wave32
// shuffle reduction; output multiplied by mp afterwards, matching reference).
// ---------------------------------------------------------------------------
__global__ __launch_bounds__(256) void softmax_kernel(
    float* __restrict__ attn, const float* __restrict__ mp)
{
  const int p = blockIdx.x * 8 + (threadIdx.x >> 5);
  const int lane = threadIdx.x & 31;
  float* row = attn + (size_t)p * NPIX;

  float m = -3.0e38f, s = 0.f;
  for (int n = lane; n < NPIX; n += 32) {
    float z = row[n] * mp[n] * 10.0f;
    float mn = fmaxf(m, z);
    s = s * __expf(m - mn) + __expf(z - mn);
    m = mn;
  }
  #pragma unroll
  for (int off = 16; off > 0; off >>= 1) {
    float m2 = __shfl_xor(m, off, 32);
    float s2 = __shfl_xor(s, off, 32);
    float mn = fmaxf(m, m2);
    s = s * __expf(m - mn) + s2 * __expf(m2 - mn);
    m = mn;
  }
  const float inv = 1.0f / s;
  for (int n = lane; n < NPIX; n += 32) {
    float g = mp[n];
    float z = row[n] * g * 10.0f;
    row[n] = __expf(z - m) * inv * g;
  }
}

// ---------------------------------------------------------------------------
// GEMM2: V[k][p] = sum_n Pbnt[k][n] * attn[p][n]
// Block tile 64(k) x 256(p), 4 waves, each wave a 64x64 WMMA tile.
// Contraction n staged in LDS chunks of 32, pitch 36.
// ---------------------------------------------------------------------------
__global__ __launch_bounds__(128) void gemm2_kernel(
    const float* __restrict__ Pbnt,  // [KDIM(k)][NPIX(n)]
    const float* __restrict__ attn,  // [NPIX(p)][NPIX(n)]
    float* __restrict__ V)           // [KDIM(k)][NPIX(p)]
{
  __shared__ float la[64][36];    // [k_local][n_local]
  __shared__ float lb[256][36];   // [p_local][n_local]

  const int tid = threadIdx.x, lane = tid & 31, wave = tid >> 5;
  const int wq = wave;            // 0..3 : p sub-tile (64)
  const int k0 = blockIdx.y * 64, q0 = blockIdx.x * 256;
  const int fr = lane & 15;
  const int fk = (lane >> 4) << 1;

  v8f acc[4][4];
  #pragma unroll
  for (int i = 0; i < 4; ++i)
    #pragma unroll
    for (int j = 0; j < 4; ++j) acc[i][j] = (v8f){};

  for (int nc = 0; nc < NPIX; nc += 32) {
    __syncthreads();
    #pragma unroll
    for (int i = 0; i < 4; ++i) {                 // A: 64k x 32n = 512 float4
      int e = tid + i * 128;
      int row = e >> 3, col4 = (e & 7) << 2;
      *(float4*)&la[row][col4] =
          *(const float4*)&Pbnt[(size_t)(k0 + row) * NPIX + nc + col4];
    }
    #pragma unroll
    for (int i = 0; i < 16; ++i) {                // B: 256p x 32n = 2048 float4
      int e = tid + i * 128;
      int row = e >> 3, col4 = (e & 7) << 2;
      *(float4*)&lb[row][col4] =
          *(const float4*)&attn[(size_t)(q0 + row) * NPIX + nc + col4];
    }
    __syncthreads();

    #pragma unroll
    for (int kk = 0; kk < 32; kk += 4) {
      v2f af[4], bf[4];
      #pragma unroll
      for (int i = 0; i < 4; ++i)
        af[i] = *(const v2f*)&la[i * 16 + fr][kk + fk];
      #pragma unroll
      for (int j = 0; j < 4; ++j)
        bf[j] = *(const v2f*)&lb[wq * 64 + j * 16 + fr][kk + fk];
      #pragma unroll
      for (int i = 0; i < 4; ++i)
        #pragma unroll
        for (int j = 0; j < 4; ++j)
          acc[i][j] = WMMA_F32(af[i], bf[j], acc[i][j]);
    }
  }

  const int dr = (lane >> 4) << 3, dc = lane & 15;
  #pragma unroll
  for (int i = 0; i < 4; ++i)
    #pragma unroll
    for (int j = 0; j < 4; ++j) {
      float* dst = V + (size_t)(k0 + i * 16 + dr) * NPIX
                     + q0 + wq * 64 + j * 16 + dc;
      #pragma unroll
      for (int r = 0; r < 8; ++r) dst[(size_t)r * NPIX] = acc[i][j][r];
    }
}

// ---------------------------------------------------------------------------
// Fold (col2im): out[c,y,x] = sum_{dy,dx} V[(c,dy,dx)][(y+1-dy, x+1-dx)]
// ---------------------------------------------------------------------------
__global__ __launch_bounds__(256) void fold_kernel(
    const float* __restrict__ V, float* __restrict__ out)
{
  int t = blockIdx.x * 256 + threadIdx.x;        // over C*H*W
  if (t >= CCH * NPIX) return;
  int c = t / NPIX, p = t % NPIX;
  int y = p / HW, x = p % HW;
  float acc = 0.f;
  #pragma unroll
  for (int dy = 0; dy < 3; ++dy) {
    int yy = y + 1 - dy;
    if ((unsigned)yy >= (unsigned)HW) continue;
    #pragma unroll
    for (int dx = 0; dx < 3; ++dx) {
      int xx = x + 1 - dx;
      if ((unsigned)xx >= (unsigned)HW) continue;
      acc += V[(size_t)(c * 9 + dy * 3 + dx) * NPIX + yy * HW + xx];
    }
  }
  out[t] = acc;
}

// ---------------------------------------------------------------------------
// Host launcher: per-batch pipeline (workspace reused across batches, ~43 MB)
// ---------------------------------------------------------------------------
extern "C" void kernel_launch(void* const* d_in, const int* in_sizes, int n_in,
                              void* d_out, int out_size, void* d_ws, size_t ws_size,
                              hipStream_t stream) {
  const float* fg   = (const float*)d_in[0];   // [4,64,48,48]
  const float* bg   = (const float*)d_in[1];   // [4,64,48,48]
  const float* mask = (const float*)d_in[2];   // [4,1,48,48]
  float* out = (float*)d_out;

  float* ws = (float*)d_ws;
  const size_t PK = (size_t)NPIX * KDIM;       // 1,327,104
  const size_t PN = (size_t)NPIX * NPIX;       // 5,308,416
  float* mp   = ws;
  float* rn   = ws + NPIX;
  float* Pf   = ws + 2 * NPIX;
  float* Pbn  = Pf + PK;
  float* Pbnt = Pbn + PK;
  float* attn = Pbnt + PK;
  float* V    = attn + PN;

  for (int b = 0; b < 4; ++b) {
    const float* fgb = fg + (size_t)b * CCH * NPIX;
    const float* bgb = bg + (size_t)b * CCH * NPIX;
    const float* mb  = mask + (size_t)b * NPIX;
    float* outb = out + (size_t)b * CCH * NPIX;

    prep_norm_kernel<<<(NPIX + 255) / 256, 256, 0, stream>>>(bgb, mb, mp, rn);
    prep_unfold_kernel<<<(NPIX * KDIM) / 256, 256, 0, stream>>>(fgb, bgb, rn, Pf, Pbn);
    prep_unfold_t_kernel<<<(NPIX * KDIM) / 256, 256, 0, stream>>>(bgb, rn, Pbnt);

    dim3 g1(NPIX / 256, NPIX / 128);           // (9, 18)
    gemm1_kernel<<<g1, 256, 0, stream>>>(Pf, Pbn, attn);

    softmax_kernel<<<NPIX / 8, 256, 0, stream>>>(attn, mp);

    dim3 g2(NPIX / 256, KDIM / 64);            // (9, 9)
    gemm2_kernel<<<g2, 128, 0, stream>>>(Pbnt, attn, V);

    fold_kernel<<<(CCH * NPIX) / 256, 256, 0, stream>>>(V, outb);
  }
}